// ImportanceScoreArch_12953621365187
// MI455X (gfx1250) — compile-verified
//
#include <hip/hip_runtime.h>
#include <hip/hip_bf16.h>

typedef float v2f __attribute__((ext_vector_type(2)));
typedef float v8f __attribute__((ext_vector_type(8)));

#define D_IN 512
#define D_H  128

// ---------------------------------------------------------------------------
// Weight packing: W[K x 128] -> fragment-ordered Wp so that the B-fragment of
// (k-quad kq, n-tile t) for lane L is the float2 at Wp[((kq*8 + t)*32 + L)].
// Per CDNA5 ISA 7.12.2 B(4x16) layout: lane L holds N=L&15, K=(L>>4)*2 (+1).
// ---------------------------------------------------------------------------
__global__ __launch_bounds__(256)
void pack_w_kernel(const float* __restrict__ W, float* __restrict__ Wp, int K)
{
    int idx = blockIdx.x * 256 + threadIdx.x;     // one float2 per thread
    int total = (K / 4) * 256;                    // (K/4) quads * 8 tiles * 32 lanes
    if (idx < total) {
        int lane = idx & 31;
        int t    = (idx >> 5) & 7;
        int kq   = idx >> 8;
        int kk   = kq * 4 + (lane >> 4) * 2;
        int col  = t * 16 + (lane & 15);
        Wp[2 * idx + 0] = W[(size_t)kk * D_H + col];
        Wp[2 * idx + 1] = W[(size_t)(kk + 1) * D_H + col];
    }
}

// ---------------------------------------------------------------------------
// WMMA fp32 GEMM:  out[n x 128] = act(A[n x K] @ W + b), W pre-packed.
// Block = 256 threads = 8 waves; block tile = 256 rows; wave tile = 32 rows
// x 128 cols (16 v8f accumulators). Per K-step: 2 A-b64 + 8 B-b64 -> 16 WMMA.
// ---------------------------------------------------------------------------
template<int KDIM, bool RELU>
__global__ __launch_bounds__(256)
void gemm_wmma_f32(const float* __restrict__ A, const float* __restrict__ Wp,
                   const float* __restrict__ bias, float* __restrict__ out,
                   int nrows)
{
    const int lane  = threadIdx.x & 31;
    const int wave  = threadIdx.x >> 5;
    const int mbase = blockIdx.x * 256 + wave * 32;
    const int mld0  = min(mbase + (lane & 15), nrows - 1);
    const int mld1  = min(mbase + 16 + (lane & 15), nrows - 1);
    const int khalf = (lane >> 4) * 2;
    const int ncol  = lane & 15;

    const v2f* __restrict__ wp = (const v2f*)Wp;

    v8f acc0[8] = {};
    v8f acc1[8] = {};

    for (int k = 0; k < KDIM; k += 4) {
        const float* a0p = A + (size_t)mld0 * KDIM + (k + khalf);
        const float* a1p = A + (size_t)mld1 * KDIM + (k + khalf);
        v2f a0; a0.x = a0p[0]; a0.y = a0p[1];
        v2f a1; a1.x = a1p[0]; a1.y = a1p[1];
        const v2f* wq = wp + ((size_t)(k >> 2) * 8) * 32 + lane;
        #pragma unroll
        for (int t = 0; t < 8; ++t) {
            v2f b = wq[t * 32];
            acc0[t] = __builtin_amdgcn_wmma_f32_16x16x4_f32(
                false, a0, false, b, (short)0, acc0[t], false, false);
            acc1[t] = __builtin_amdgcn_wmma_f32_16x16x4_f32(
                false, a1, false, b, (short)0, acc1[t], false, false);
        }
    }

    const int rbase = (lane >> 4) * 8;
    #pragma unroll
    for (int t = 0; t < 8; ++t) {
        float bv = bias[t * 16 + ncol];
        #pragma unroll
        for (int r = 0; r < 8; ++r) {
            int row0 = mbase + rbase + r;
            if (row0 < nrows) {
                float v = acc0[t][r] + bv;
                if (RELU) v = fmaxf(v, 0.f);
                out[(size_t)row0 * D_H + t * 16 + ncol] = v;
            }
            int row1 = mbase + 16 + rbase + r;
            if (row1 < nrows) {
                float v = acc1[t][r] + bv;
                if (RELU) v = fmaxf(v, 0.f);
                out[(size_t)row1 * D_H + t * 16 + ncol] = v;
            }
        }
    }
}

// ---------------------------------------------------------------------------
// Fused GIN MLP:  out = relu((x + agg) @ W1 + b1) @ W2 + b2   (weights packed,
// in-place safe: each block only touches its own 256 rows). Intermediate h in
// a 128 KB LDS tile; phase 2 reloads each wave's own rows as ds_load_b64.
// ---------------------------------------------------------------------------
__global__ __launch_bounds__(256)
void gin_mlp_wmma(const float* __restrict__ x, const float* __restrict__ agg,
                  const float* __restrict__ W1p, const float* __restrict__ b1,
                  const float* __restrict__ W2p, const float* __restrict__ b2,
                  float* __restrict__ out, int nrows)
{
    __shared__ float hbuf[256 * 128];   // 128 KB of the 320 KB WGP LDS

    const int lane  = threadIdx.x & 31;
    const int wave  = threadIdx.x >> 5;
    const int mbase = blockIdx.x * 256 + wave * 32;
    const int mld0  = min(mbase + (lane & 15), nrows - 1);
    const int mld1  = min(mbase + 16 + (lane & 15), nrows - 1);
    const int khalf = (lane >> 4) * 2;
    const int ncol  = lane & 15;
    const int rbase = (lane >> 4) * 8;

    const v2f* __restrict__ w1 = (const v2f*)W1p;
    const v2f* __restrict__ w2 = (const v2f*)W2p;

    // ---- phase 1: h = relu((x+agg) @ W1 + b1) -> LDS ----------------------
    v8f acc0[8] = {};
    v8f acc1[8] = {};
    for (int k = 0; k < D_H; k += 4) {
        const float* x0 = x   + (size_t)mld0 * D_H + (k + khalf);
        const float* g0 = agg + (size_t)mld0 * D_H + (k + khalf);
        const float* x1 = x   + (size_t)mld1 * D_H + (k + khalf);
        const float* g1 = agg + (size_t)mld1 * D_H + (k + khalf);
        v2f a0; a0.x = x0[0] + g0[0]; a0.y = x0[1] + g0[1];
        v2f a1; a1.x = x1[0] + g1[0]; a1.y = x1[1] + g1[1];
        const v2f* wq = w1 + ((size_t)(k >> 2) * 8) * 32 + lane;
        #pragma unroll
        for (int t = 0; t < 8; ++t) {
            v2f b = wq[t * 32];
            acc0[t] = __builtin_amdgcn_wmma_f32_16x16x4_f32(
                false, a0, false, b, (short)0, acc0[t], false, false);
            acc1[t] = __builtin_amdgcn_wmma_f32_16x16x4_f32(
                false, a1, false, b, (short)0, acc1[t], false, false);
        }
    }
    #pragma unroll
    for (int t = 0; t < 8; ++t) {
        float bv = b1[t * 16 + ncol];
        #pragma unroll
        for (int r = 0; r < 8; ++r) {
            int lr0 = wave * 32 + rbase + r;
            hbuf[lr0 * 128 + t * 16 + ncol]        = fmaxf(acc0[t][r] + bv, 0.f);
            hbuf[(lr0 + 16) * 128 + t * 16 + ncol] = fmaxf(acc1[t][r] + bv, 0.f);
        }
    }
    __syncthreads();

    // ---- phase 2: out = h @ W2 + b2 ---------------------------------------
    #pragma unroll
    for (int t = 0; t < 8; ++t) { acc0[t] = (v8f)0.f; acc1[t] = (v8f)0.f; }
    const float* h0 = hbuf + (size_t)(wave * 32 + (lane & 15)) * 128;
    const float* h1 = h0 + 16 * 128;
    for (int k = 0; k < D_H; k += 4) {
        v2f a0; a0.x = h0[k + khalf]; a0.y = h0[k + khalf + 1];
        v2f a1; a1.x = h1[k + khalf]; a1.y = h1[k + khalf + 1];
        const v2f* wq = w2 + ((size_t)(k >> 2) * 8) * 32 + lane;
        #pragma unroll
        for (int t = 0; t < 8; ++t) {
            v2f b = wq[t * 32];
            acc0[t] = __builtin_amdgcn_wmma_f32_16x16x4_f32(
                false, a0, false, b, (short)0, acc0[t], false, false);
            acc1[t] = __builtin_amdgcn_wmma_f32_16x16x4_f32(
                false, a1, false, b, (short)0, acc1[t], false, false);
        }
    }
    #pragma unroll
    for (int t = 0; t < 8; ++t) {
        float bv = b2[t * 16 + ncol];
        #pragma unroll
        for (int r = 0; r < 8; ++r) {
            int row0 = mbase + rbase + r;
            if (row0 < nrows)
                out[(size_t)row0 * D_H + t * 16 + ncol] = acc0[t][r] + bv;
            int row1 = mbase + 16 + rbase + r;
            if (row1 < nrows)
                out[(size_t)row1 * D_H + t * 16 + ncol] = acc1[t][r] + bv;
        }
    }
}

// ---------------------------------------------------------------------------
// agg[dst] += x[src] over E edges; 2 edges per 256-thread block, one lane per
// feature. Rows are L2-resident (25.6 MB); native f32 atomics.
// ---------------------------------------------------------------------------
__global__ __launch_bounds__(256)
void scatter_add_kernel(const float* __restrict__ x, const int* __restrict__ ei,
                        float* __restrict__ agg, int E)
{
    int e = blockIdx.x * 2 + (threadIdx.x >> 7);
    int f = threadIdx.x & 127;
    if (e < E) {
        int s = ei[e];
        int d = ei[E + e];
        unsafeAtomicAdd(&agg[(size_t)d * D_H + f], x[(size_t)s * D_H + f]);
    }
}

__global__ void zero_kernel(float* __restrict__ p, int n)
{
    int i = blockIdx.x * blockDim.x + threadIdx.x;
    int stride = gridDim.x * blockDim.x;
    for (; i < n; i += stride) p[i] = 0.f;
}

// ---------------------------------------------------------------------------
// Tail: z[i] = dot(x[i,:], tW) + tb; deterministic per-block partial sums of
// z and z^2 for BatchNorm. One wave handles 32 rows; each row is one fully
// coalesced float4 read across 32 lanes + __shfl_xor reduce (wave32).
// ---------------------------------------------------------------------------
__global__ __launch_bounds__(256)
void tail_kernel(const float* __restrict__ x, const float* __restrict__ tW,
                 const float* __restrict__ tb, float* __restrict__ z,
                 float* __restrict__ part, int nrows)
{
    __shared__ float sW[128];
    __shared__ float ssum[8], ssq[8];
    const int tid  = threadIdx.x;
    const int lane = tid & 31;
    const int wave = tid >> 5;
    if (tid < 128) sW[tid] = tW[tid];
    __syncthreads();

    float w0 = sW[lane * 4 + 0], w1 = sW[lane * 4 + 1];
    float w2 = sW[lane * 4 + 2], w3 = sW[lane * 4 + 3];
    float tbv = tb[0];

    float lsum = 0.f, lsq = 0.f;
    int rowBase = blockIdx.x * 256 + wave * 32;
    for (int i = 0; i < 32; ++i) {
        int row = rowBase + i;                 // wave-uniform
        float p = 0.f;
        if (row < nrows) {
            const float4 xv = ((const float4*)(x + (size_t)row * D_H))[lane];
            p = xv.x * w0 + xv.y * w1 + xv.z * w2 + xv.w * w3;
        }
        #pragma unroll
        for (int off = 16; off > 0; off >>= 1)
            p += __shfl_xor(p, off, 32);
        if (row < nrows && lane == 0) {
            float zv = p + tbv;
            z[row] = zv;
            lsum += zv;
            lsq  += zv * zv;
        }
    }
    if (lane == 0) { ssum[wave] = lsum; ssq[wave] = lsq; }
    __syncthreads();
    if (tid == 0) {
        float a = 0.f, b = 0.f;
        #pragma unroll
        for (int w = 0; w < 8; ++w) { a += ssum[w]; b += ssq[w]; }
        part[blockIdx.x * 2 + 0] = a;
        part[blockIdx.x * 2 + 1] = b;
    }
}

__global__ __launch_bounds__(256)
void stats_kernel(const float* __restrict__ part, float* __restrict__ stats,
                  int nparts, float inv_n)
{
    __shared__ float bs[256], bq[256];
    float s = 0.f, q = 0.f;
    for (int i = threadIdx.x; i < nparts; i += 256) {
        s += part[2 * i + 0];
        q += part[2 * i + 1];
    }
    bs[threadIdx.x] = s; bq[threadIdx.x] = q;
    __syncthreads();
    for (int off = 128; off > 0; off >>= 1) {
        if (threadIdx.x < off) {
            bs[threadIdx.x] += bs[threadIdx.x + off];
            bq[threadIdx.x] += bq[threadIdx.x + off];
        }
        __syncthreads();
    }
    if (threadIdx.x == 0) {
        float mu  = bs[0] * inv_n;
        float var = bq[0] * inv_n - mu * mu;
        stats[0] = mu;
        stats[1] = rsqrtf(var + 1e-5f);
    }
}

__global__ void norm_kernel(const float* __restrict__ z,
                            const float* __restrict__ stats,
                            const float* __restrict__ gamma,
                            const float* __restrict__ beta,
                            float* __restrict__ out, int n)
{
    int i = blockIdx.x * blockDim.x + threadIdx.x;
    if (i < n)
        out[i] = (z[i] - stats[0]) * stats[1] * gamma[0] + beta[0];
}

// ---------------------------------------------------------------------------
extern "C" void kernel_launch(void* const* d_in, const int* in_sizes, int n_in,
                              void* d_out, int out_size, void* d_ws, size_t ws_size,
                              hipStream_t stream)
{
    const float* feature = (const float*)d_in[0];
    const int*   ei      = (const int*)  d_in[1];
    const float* head_W  = (const float*)d_in[2];
    const float* head_b  = (const float*)d_in[3];
    const float* gin_W1  = (const float*)d_in[4];
    const float* gin_b1  = (const float*)d_in[5];
    const float* lin_W1  = (const float*)d_in[6];
    const float* lin_b1  = (const float*)d_in[7];
    const float* gin_W2  = (const float*)d_in[8];
    const float* gin_b2  = (const float*)d_in[9];
    const float* lin_W2  = (const float*)d_in[10];
    const float* lin_b2  = (const float*)d_in[11];
    const float* tail_W  = (const float*)d_in[12];
    const float* tail_b  = (const float*)d_in[13];
    const float* bn_g    = (const float*)d_in[14];
    const float* bn_b    = (const float*)d_in[15];

    const int N = in_sizes[0] / D_IN;
    const int E = in_sizes[1] / 2;

    // Workspace layout (floats):
    //   x | agg | z | partials | stats | packed weights
    float* bufA   = (float*)d_ws;                  // N*128 (x, in place)
    float* bufB   = bufA + (size_t)N * D_H;        // N*128 (agg)
    float* zbuf   = bufB + (size_t)N * D_H;        // N
    float* part   = zbuf + N;                      // 2 * tailBlocks (pad 8192)
    float* stats  = part + 2 * 4096;               // 2 (pad 16)
    float* wpHead = stats + 16;                    // (512/4)*256*2 = 65536
    float* wpG1   = wpHead + 65536;                // (128/4)*256*2 = 16384
    float* wpL1   = wpG1 + 16384;
    float* wpG2   = wpL1 + 16384;
    float* wpL2   = wpG2 + 16384;

    const int gemmBlocks = (N + 255) / 256;
    const int tailBlocks = (N + 255) / 256;

    // Pack weights into fragment order (tiny, once per call).
    pack_w_kernel<<<(D_IN / 4 * 256 + 255) / 256, 256, 0, stream>>>(head_W, wpHead, D_IN);
    pack_w_kernel<<<(D_H / 4 * 256 + 255) / 256, 256, 0, stream>>>(gin_W1, wpG1, D_H);
    pack_w_kernel<<<(D_H / 4 * 256 + 255) / 256, 256, 0, stream>>>(lin_W1, wpL1, D_H);
    pack_w_kernel<<<(D_H / 4 * 256 + 255) / 256, 256, 0, stream>>>(gin_W2, wpG2, D_H);
    pack_w_kernel<<<(D_H / 4 * 256 + 255) / 256, 256, 0, stream>>>(lin_W2, wpL2, D_H);

    // head: x = relu(feature @ head_W + head_b)
    gemm_wmma_f32<D_IN, true><<<gemmBlocks, 256, 0, stream>>>(
        feature, wpHead, head_b, bufA, N);

    for (int layer = 0; layer < 2; ++layer) {
        const float* W1p = layer ? wpG2 : wpG1;
        const float* B1  = layer ? gin_b2 : gin_b1;
        const float* W2p = layer ? wpL2 : wpL1;
        const float* B2  = layer ? lin_b2 : lin_b1;

        zero_kernel<<<2048, 256, 0, stream>>>(bufB, N * D_H);
        scatter_add_kernel<<<(E + 1) / 2, 256, 0, stream>>>(bufA, ei, bufB, E);
        gin_mlp_wmma<<<gemmBlocks, 256, 0, stream>>>(
            bufA, bufB, W1p, B1, W2p, B2, bufA, N);
    }

    tail_kernel<<<tailBlocks, 256, 0, stream>>>(bufA, tail_W, tail_b, zbuf, part, N);
    stats_kernel<<<1, 256, 0, stream>>>(part, stats, tailBlocks, 1.0f / (float)N);
    norm_kernel<<<(N + 255) / 256, 256, 0, stream>>>(zbuf, stats, bn_g, bn_b,
                                                     (float*)d_out, N);
}